// DCDicL_30210799960323
// MI455X (gfx1250) — compile-verified
//
#include <hip/hip_runtime.h>
#include <math.h>

// ---------------------------------------------------------------------------
// DCDicL step for MI455X (gfx1250, wave32, WMMA).
// Compute-bound: 81 Gram GEMMs (43.5 GFLOP) -> v_wmma_f32_16x16x4_f32.
// Everything fits in 192MB L2; HBM traffic ~40MB total.
// ---------------------------------------------------------------------------

typedef __attribute__((ext_vector_type(2))) float v2f;
typedef __attribute__((ext_vector_type(8))) float v8f;

#define NBATCH 4
#define C_IN   64
#define HH     128
#define WW     128
#define WF     65      // rfft bins along W
#define NOFF   81      // 9x9 correlation offsets
#define MDIM   1600    // C_IN * 5 * 5
#define NB     16
#define NBLK   100     // MDIM / NB

// workspace byte offsets (phase1 spectra overlap phase2 Q; lifetimes disjoint)
#define OFF_A    0UL          // 256*128*65*8 = 17,039,360  (row-FFT x / Xn)
#define OFF_B    17039360UL   // X spectrum / inv-col temp
#define OFF_C    34078720UL   // D (OTF)
#define OFF_TY   51118080UL   // y row-FFT (4*128*65*8)
#define OFF_Y    51384320UL   // Y spectrum
#define OFF_Q    0UL          // 4*1600*1600*4 = 40,960,000
#define OFF_CORR 41000192UL   // 4*64*64*81*4  = 5,308,416
#define OFF_PM   46308864UL   // 4*1600*4

__device__ __forceinline__ int bitrev7(int j) { return (int)(__brev((unsigned)j) >> 25); }

// ---------------- forward FFT: rows (real input -> 65 complex bins) --------
__global__ void __launch_bounds__(256) k_fft_rows_fwd(const float* __restrict__ img,
                                                      float2* __restrict__ T) {
  int blk  = blockIdx.x;
  int imgId = blk >> 5;
  int row   = ((blk & 31) << 2) + (threadIdx.x >> 6);
  int slot  = threadIdx.x >> 6;
  int t     = threadIdx.x & 63;
  __shared__ float re[4][128], im[4][128];
  const float* src = img + ((size_t)imgId * HH + row) * WW;
  for (int e = 0; e < 2; ++e) {
    int j = t + (e << 6);
    re[slot][j] = src[bitrev7(j)];
    im[slot][j] = 0.0f;
  }
  __syncthreads();
  for (int len = 2; len <= 128; len <<= 1) {
    int half = len >> 1;
    int g = t / half, k = t % half;
    int i0 = g * len + k, i1 = i0 + half;
    float s, c;
    __sincosf(-6.2831853071795864f * (float)k / (float)len, &s, &c);
    float xr = re[slot][i1], xi = im[slot][i1];
    float tr = c * xr - s * xi, ti = c * xi + s * xr;
    float ur = re[slot][i0], ui = im[slot][i0];
    re[slot][i0] = ur + tr; im[slot][i0] = ui + ti;
    re[slot][i1] = ur - tr; im[slot][i1] = ui - ti;
    __syncthreads();
  }
  for (int wf = t; wf < WF; wf += 64)
    T[((size_t)imgId * HH + row) * WF + wf] = make_float2(re[slot][wf], im[slot][wf]);
}

// ---------------- column FFT (fwd or inv), 128-pt complex ------------------
__global__ void __launch_bounds__(256) k_fft_cols(const float2* __restrict__ T,
                                                  float2* __restrict__ O, int inverse) {
  int blk   = blockIdx.x;
  int imgId = blk / 17;
  int col   = (blk % 17) * 4 + (threadIdx.x >> 6);
  int slot  = threadIdx.x >> 6;
  int t     = threadIdx.x & 63;
  bool act  = (col < WF);
  __shared__ float re[4][128], im[4][128];
  for (int e = 0; e < 2; ++e) {
    int j = t + (e << 6);
    float2 v = make_float2(0.f, 0.f);
    if (act) v = T[((size_t)imgId * HH + bitrev7(j)) * WF + col];
    re[slot][j] = v.x; im[slot][j] = v.y;
  }
  __syncthreads();
  float sgn = inverse ? 1.0f : -1.0f;
  for (int len = 2; len <= 128; len <<= 1) {
    int half = len >> 1;
    int g = t / half, k = t % half;
    int i0 = g * len + k, i1 = i0 + half;
    float s, c;
    __sincosf(sgn * 6.2831853071795864f * (float)k / (float)len, &s, &c);
    float xr = re[slot][i1], xi = im[slot][i1];
    float tr = c * xr - s * xi, ti = c * xi + s * xr;
    float ur = re[slot][i0], ui = im[slot][i0];
    re[slot][i0] = ur + tr; im[slot][i0] = ui + ti;
    re[slot][i1] = ur - tr; im[slot][i1] = ui - ti;
    __syncthreads();
  }
  float scale = inverse ? (1.0f / 128.0f) : 1.0f;
  if (act)
    for (int e = 0; e < 2; ++e) {
      int j = t + (e << 6);
      O[((size_t)imgId * HH + j) * WF + col] = make_float2(re[slot][j] * scale, im[slot][j] * scale);
    }
}

// ---------------- inverse row FFT: Hermitian 65 -> 128 reals ---------------
__global__ void __launch_bounds__(256) k_ifft_rows(const float2* __restrict__ T,
                                                   float* __restrict__ out) {
  int blk  = blockIdx.x;
  int imgId = blk >> 5;
  int row   = ((blk & 31) << 2) + (threadIdx.x >> 6);
  int slot  = threadIdx.x >> 6;
  int t     = threadIdx.x & 63;
  __shared__ float re[4][128], im[4][128];
  const float2* src = T + ((size_t)imgId * HH + row) * WF;
  for (int e = 0; e < 2; ++e) {
    int j = t + (e << 6);
    int idx = bitrev7(j);
    float2 v;
    if (idx <= 64) v = src[idx];
    else { float2 u = src[128 - idx]; v = make_float2(u.x, -u.y); }
    re[slot][j] = v.x; im[slot][j] = v.y;
  }
  __syncthreads();
  for (int len = 2; len <= 128; len <<= 1) {
    int half = len >> 1;
    int g = t / half, k = t % half;
    int i0 = g * len + k, i1 = i0 + half;
    float s, c;
    __sincosf(6.2831853071795864f * (float)k / (float)len, &s, &c);
    float xr = re[slot][i1], xi = im[slot][i1];
    float tr = c * xr - s * xi, ti = c * xi + s * xr;
    float ur = re[slot][i0], ui = im[slot][i0];
    re[slot][i0] = ur + tr; im[slot][i0] = ui + ti;
    re[slot][i1] = ur - tr; im[slot][i1] = ui - ti;
    __syncthreads();
  }
  float* dst = out + ((size_t)imgId * HH + row) * WW;
  for (int e = 0; e < 2; ++e) {
    int j = t + (e << 6);
    dst[j] = re[slot][j] * (1.0f / 128.0f);
  }
}

// ---------------- OTF via direct 5x5 DFT (psf rolled by -2,-2) -------------
__global__ void k_p2o(const float* __restrict__ d, float2* __restrict__ D) {
  int gid = blockIdx.x * 256 + threadIdx.x;
  if (gid >= NBATCH * C_IN * HH * WF) return;
  int wf = gid % WF;
  int u  = (gid / WF) % HH;
  int cc = (gid / (WF * HH)) % C_IN;
  int n  = gid / (WF * HH * C_IN);
  const float* psf = d + ((size_t)(n * C_IN + cc)) * 25;
  float sr = 0.f, si = 0.f;
  const float w0 = -6.2831853071795864f / 128.0f;
  for (int i = 0; i < 5; ++i)
    for (int j = 0; j < 5; ++j) {
      float ph = w0 * (float)(u * (i - 2) + wf * (j - 2));
      float s, c; __sincosf(ph, &s, &c);
      float p = psf[i * 5 + j];
      sr += p * c; si += p * s;
    }
  D[gid] = make_float2(sr, si);
}

// ---------------- closed-form x-update per frequency bin -------------------
__global__ void k_solve_fft(const float2* __restrict__ X, const float2* __restrict__ D,
                            const float2* __restrict__ Y, const float* __restrict__ alpha_x,
                            float2* __restrict__ Xn) {
  int gid = blockIdx.x * 256 + threadIdx.x;
  if (gid >= NBATCH * HH * WF) return;
  int wf = gid % WF;
  int h  = (gid / WF) % HH;
  int n  = gid / (WF * HH);
  float a = alpha_x[n] / (float)C_IN;
  float2 Yv = Y[((size_t)n * HH + h) * WF + wf];
  size_t base = (((size_t)n * C_IN) * HH + h) * WF + wf;
  const size_t cs = (size_t)HH * WF;
  float nr = 0.f, ni = 0.f, sden = 0.f;
  for (int c = 0; c < C_IN; ++c) {
    float2 Dv = D[base + c * cs];
    float2 Xv = X[base + c * cs];
    float zr = Yv.x * Dv.x - Yv.y * Dv.y + a * Xv.x;
    float zi = Yv.x * Dv.y + Yv.y * Dv.x + a * Xv.y;
    nr += Dv.x * zr + Dv.y * zi;
    ni += Dv.x * zi - Dv.y * zr;
    sden += Dv.x * Dv.x + Dv.y * Dv.y;
  }
  float den = a * sden + a * a;
  float rr = nr / den, ri = ni / den, inva = 1.0f / a;
  for (int c = 0; c < C_IN; ++c) {
    float2 Dv = D[base + c * cs];
    float2 Xv = X[base + c * cs];
    float zr = Yv.x * Dv.x - Yv.y * Dv.y + a * Xv.x;
    float zi = Yv.x * Dv.y + Yv.y * Dv.x + a * Xv.y;
    Xn[base + c * cs] = make_float2(zr * inva - (Dv.x * rr - Dv.y * ri),
                                    zi * inva - (Dv.x * ri + Dv.y * rr));
  }
}

// ---------------- 64x64 Gram per (batch, offset) via f32 WMMA --------------
// corr[n,c1,c2,dy,dx] = sum_{h,w} x[c1,h+oy,w+ox] * x[c2,h,w]  (valid region)
#define GK 96
__global__ void __launch_bounds__(256) k_gram_wmma(const float* __restrict__ x,
                                                   float* __restrict__ corr) {
  int n   = blockIdx.x / NOFF;
  int off = blockIdx.x % NOFF;
  int oy = off / 9 - 4, ox = off % 9 - 4;
  int h0 = oy < 0 ? -oy : 0;
  int w0 = ox < 0 ? -ox : 0;
  int Hv = 128 - (oy < 0 ? -oy : oy);
  int Wv = 128 - (ox < 0 ? -ox : ox);
  int Pv = Hv * Wv;
  __shared__ float As[64][GK + 1], Bs[64][GK + 1];
  int tid = threadIdx.x, wave = tid >> 5, lane = tid & 31;
  int tA = wave * 2;              // two 16x16 tiles per wave, same M-row
  int tm = tA >> 2, tn0 = tA & 3, tn1 = (tA + 1) & 3;
  v8f acc0 = {}; v8f acc1 = {};
  const float* xb = x + (size_t)n * C_IN * HH * WW;
  int mr = lane & 15;
  int ksel = (lane >> 4) << 1;
  for (int p0 = 0; p0 < Pv; p0 += GK) {
    for (int idx = tid; idx < 64 * GK; idx += 256) {
      int c = idx / GK, k = idx - c * GK;
      int p = p0 + k;
      float av = 0.f, bv = 0.f;
      if (p < Pv) {
        int h = p / Wv, w = p - h * Wv;
        av = xb[(size_t)c * HH * WW + (h0 + h + oy) * WW + (w0 + w + ox)];
        bv = xb[(size_t)c * HH * WW + (h0 + h) * WW + (w0 + w)];
      }
      As[c][k] = av; Bs[c][k] = bv;
    }
    __syncthreads();
    for (int k = 0; k < GK; k += 4) {
      int kk = k + ksel;
      v2f a, b0, b1;
      a.x  = As[tm * 16 + mr][kk];  a.y  = As[tm * 16 + mr][kk + 1];
      b0.x = Bs[tn0 * 16 + mr][kk]; b0.y = Bs[tn0 * 16 + mr][kk + 1];
      b1.x = Bs[tn1 * 16 + mr][kk]; b1.y = Bs[tn1 * 16 + mr][kk + 1];
      acc0 = __builtin_amdgcn_wmma_f32_16x16x4_f32(false, a, false, b0, (short)0, acc0, false, false);
      acc1 = __builtin_amdgcn_wmma_f32_16x16x4_f32(false, a, false, b1, (short)0, acc1, false, false);
    }
    __syncthreads();
  }
  int nn0 = tn0 * 16 + mr, nn1 = tn1 * 16 + mr;
  int mbase = tm * 16 + ((lane >> 4) << 3);
  for (int r = 0; r < 8; ++r) {
    int c1 = mbase + r;
    corr[(((size_t)n * 64 + c1) * 64 + nn0) * NOFF + off] = acc0[r];
    corr[(((size_t)n * 64 + c1) * 64 + nn1) * NOFF + off] = acc1[r];
  }
}

// ---------------- assemble Q (1600x1600) + ridge ---------------------------
__global__ void k_assemble_Q(const float* __restrict__ corr, const float* __restrict__ alpha_d,
                             const float* __restrict__ reg, float* __restrict__ Q) {
  int gid = blockIdx.x * 256 + threadIdx.x;
  if (gid >= NBATCH * MDIM * MDIM) return;
  int mc = gid % MDIM;
  int mrw = (gid / MDIM) % MDIM;
  int n  = gid / (MDIM * MDIM);
  int c2 = mrw / 25, ar = (mrw / 5) % 5, cd = mrw % 5;
  int c1 = mc / 25,  br = (mc / 5) % 5,  d2 = mc % 5;
  float v = corr[(((size_t)n * 64 + c1) * 64 + c2) * NOFF + (4 + br - ar) * 9 + (4 + d2 - cd)];
  if (mrw == mc) v += alpha_d[n] * ((float)(HH * WW) * reg[0] / (float)MDIM);
  Q[gid] = v;
}

// ---------------- Pm = xty + ridge * d_old ---------------------------------
__global__ void __launch_bounds__(256) k_pm(const float* __restrict__ xnew,
                                            const float* __restrict__ y,
                                            const float* __restrict__ dold,
                                            const float* __restrict__ alpha_d,
                                            const float* __restrict__ reg,
                                            float* __restrict__ Pm) {
  int blk = blockIdx.x;
  int ab  = blk % 25;
  int ci  = (blk / 25) % C_IN;
  int n   = blk / (25 * C_IN);
  int qy = ab / 5 - 2, qx = ab % 5 - 2;
  int h0 = qy < 0 ? -qy : 0, w0 = qx < 0 ? -qx : 0;
  int Hv = 128 - (qy < 0 ? -qy : qy), Wv = 128 - (qx < 0 ? -qx : qx);
  const float* xp = xnew + ((size_t)n * C_IN + ci) * HH * WW;
  const float* yp = y + (size_t)n * HH * WW;
  float s = 0.f;
  for (int p = threadIdx.x; p < Hv * Wv; p += 256) {
    int h = p / Wv, w = p - h * Wv;
    s += xp[(h0 + h + qy) * WW + (w0 + w + qx)] * yp[(h0 + h) * WW + (w0 + w)];
  }
  __shared__ float red[256];
  red[threadIdx.x] = s;
  __syncthreads();
  for (int st = 128; st > 0; st >>= 1) {
    if (threadIdx.x < st) red[threadIdx.x] += red[threadIdx.x + st];
    __syncthreads();
  }
  if (threadIdx.x == 0) {
    float areg = alpha_d[n] * ((float)(HH * WW) * reg[0] / (float)MDIM);
    Pm[n * MDIM + ci * 25 + ab] = red[0] + areg * dold[((size_t)n * C_IN + ci) * 25 + ab];
  }
}

// ---------------- blocked Cholesky (NB=16, WMMA SYRK) + triangular solves --
__global__ void __launch_bounds__(1024) k_cholesky_solve(float* __restrict__ Q,
                                                         const float* __restrict__ Pm,
                                                         float* __restrict__ dout) {
  int n = blockIdx.x;
  float* Qn = Q + (size_t)n * MDIM * MDIM;
  int tid = threadIdx.x, wave = tid >> 5, lane = tid & 31;
  __shared__ float Lkk[NB][NB + 1];
  __shared__ float pv[MDIM];

  for (int kb = 0; kb < NBLK; ++kb) {
    int j0 = kb * NB;
    if (tid < 256) {
      int r = tid >> 4, c = tid & 15;
      Lkk[r][c] = Qn[(size_t)(j0 + r) * MDIM + j0 + c];
    }
    __syncthreads();
    for (int c = 0; c < NB; ++c) {                 // unblocked 16x16 factor
      if (tid == 0) Lkk[c][c] = sqrtf(fmaxf(Lkk[c][c], 1e-30f));
      __syncthreads();
      if (tid > c && tid < NB) Lkk[tid][c] /= Lkk[c][c];
      __syncthreads();
      if (tid < 256) {
        int r = tid >> 4, c2 = tid & 15;
        if (r > c && c2 > c && c2 <= r) Lkk[r][c2] -= Lkk[r][c] * Lkk[c2][c];
      }
      __syncthreads();
    }
    if (tid < 256) {
      int r = tid >> 4, c = tid & 15;
      if (c <= r) Qn[(size_t)(j0 + r) * MDIM + j0 + c] = Lkk[r][c];
    }
    __syncthreads();
    // panel: rows below solve against Lkk^T
    for (int i = j0 + NB + tid; i < MDIM; i += 1024) {
      float v[NB];
      float* rowp = Qn + (size_t)i * MDIM + j0;
      for (int c = 0; c < NB; ++c) v[c] = rowp[c];
      for (int c = 0; c < NB; ++c) {
        float s = v[c];
        for (int t2 = 0; t2 < c; ++t2) s -= v[t2] * Lkk[c][t2];
        v[c] = s / Lkk[c][c];
      }
      for (int c = 0; c < NB; ++c) rowp[c] = v[c];
    }
    __syncthreads();
    // trailing SYRK update via WMMA: C -= P_i * P_j^T (A negated at load)
    int nrem = NBLK - kb - 1;
    int ntiles = nrem * (nrem + 1) / 2;
    int mr = lane & 15, hi = lane >> 4;
    for (int t = wave; t < ntiles; t += 32) {
      int bi = (int)((sqrtf(8.0f * (float)t + 1.0f) - 1.0f) * 0.5f);
      while ((bi + 1) * (bi + 2) / 2 <= t) ++bi;
      while (bi * (bi + 1) / 2 > t) --bi;
      int bj = t - bi * (bi + 1) / 2;
      int i0 = j0 + NB + bi * NB;
      int i1 = j0 + NB + bj * NB;
      v8f cfrag;
      for (int r = 0; r < 8; ++r)
        cfrag[r] = Qn[(size_t)(i0 + r + hi * 8) * MDIM + i1 + mr];
      for (int k = 0; k < NB; k += 4) {
        int kk = j0 + k + hi * 2;
        v2f a, b;
        a.x = -Qn[(size_t)(i0 + mr) * MDIM + kk];
        a.y = -Qn[(size_t)(i0 + mr) * MDIM + kk + 1];
        b.x =  Qn[(size_t)(i1 + mr) * MDIM + kk];
        b.y =  Qn[(size_t)(i1 + mr) * MDIM + kk + 1];
        cfrag = __builtin_amdgcn_wmma_f32_16x16x4_f32(false, a, false, b, (short)0, cfrag, false, false);
      }
      for (int r = 0; r < 8; ++r)
        Qn[(size_t)(i0 + r + hi * 8) * MDIM + i1 + mr] = cfrag[r];
    }
    __syncthreads();
  }

  // forward solve L z = p (blocked)
  for (int i = tid; i < MDIM; i += 1024) pv[i] = Pm[n * MDIM + i];
  __syncthreads();
  for (int kb = 0; kb < NBLK; ++kb) {
    int j0 = kb * NB;
    if (tid == 0) {
      for (int c = 0; c < NB; ++c) {
        float s = pv[j0 + c];
        for (int t2 = 0; t2 < c; ++t2) s -= Qn[(size_t)(j0 + c) * MDIM + j0 + t2] * pv[j0 + t2];
        pv[j0 + c] = s / Qn[(size_t)(j0 + c) * MDIM + j0 + c];
      }
    }
    __syncthreads();
    for (int i = j0 + NB + tid; i < MDIM; i += 1024) {
      float s = 0.f;
      for (int c = 0; c < NB; ++c) s += Qn[(size_t)i * MDIM + j0 + c] * pv[j0 + c];
      pv[i] -= s;
    }
    __syncthreads();
  }
  // backward solve L^T w = z (blocked)
  for (int kb = NBLK - 1; kb >= 0; --kb) {
    int j0 = kb * NB;
    if (tid == 0) {
      for (int c = NB - 1; c >= 0; --c) {
        float s = pv[j0 + c];
        for (int t2 = c + 1; t2 < NB; ++t2) s -= Qn[(size_t)(j0 + t2) * MDIM + j0 + c] * pv[j0 + t2];
        pv[j0 + c] = s / Qn[(size_t)(j0 + c) * MDIM + j0 + c];
      }
    }
    __syncthreads();
    for (int i = tid; i < j0; i += 1024) {
      float s = 0.f;
      for (int c = 0; c < NB; ++c) s += Qn[(size_t)(j0 + c) * MDIM + i] * pv[j0 + c];
      pv[i] -= s;
    }
    __syncthreads();
  }
  for (int i = tid; i < MDIM; i += 1024)
    dout[(size_t)NBATCH * C_IN * HH * WW + n * MDIM + i] = pv[i];
}

// ---------------------------------------------------------------------------
extern "C" void kernel_launch(void* const* d_in, const int* in_sizes, int n_in,
                              void* d_out, int out_size, void* d_ws, size_t ws_size,
                              hipStream_t stream) {
  const float* x       = (const float*)d_in[0];
  const float* d       = (const float*)d_in[1];
  const float* y       = (const float*)d_in[2];
  const float* alpha_x = (const float*)d_in[3];
  const float* alpha_d = (const float*)d_in[4];
  const float* reg     = (const float*)d_in[5];
  float* out = (float*)d_out;
  char*  ws  = (char*)d_ws;

  float2* bufA  = (float2*)(ws + OFF_A);
  float2* bufB  = (float2*)(ws + OFF_B);
  float2* bufC  = (float2*)(ws + OFF_C);
  float2* bufTy = (float2*)(ws + OFF_TY);
  float2* bufY  = (float2*)(ws + OFF_Y);
  float*  Q     = (float*)(ws + OFF_Q);
  float*  corr  = (float*)(ws + OFF_CORR);
  float*  Pm    = (float*)(ws + OFF_PM);

  // x-update (FFT domain)
  k_fft_rows_fwd<<<NBATCH * C_IN * 32, 256, 0, stream>>>(x, bufA);
  k_fft_cols<<<NBATCH * C_IN * 17, 256, 0, stream>>>(bufA, bufB, 0);
  k_fft_rows_fwd<<<NBATCH * 32, 256, 0, stream>>>(y, bufTy);
  k_fft_cols<<<NBATCH * 17, 256, 0, stream>>>(bufTy, bufY, 0);
  k_p2o<<<(NBATCH * C_IN * HH * WF + 255) / 256, 256, 0, stream>>>(d, bufC);
  k_solve_fft<<<(NBATCH * HH * WF + 255) / 256, 256, 0, stream>>>(bufB, bufC, bufY, alpha_x, bufA);
  k_fft_cols<<<NBATCH * C_IN * 17, 256, 0, stream>>>(bufA, bufB, 1);
  k_ifft_rows<<<NBATCH * C_IN * 32, 256, 0, stream>>>(bufB, out);

  // d-update (Gram via WMMA, ridge LS, Cholesky)
  k_gram_wmma<<<NBATCH * NOFF, 256, 0, stream>>>(out, corr);
  k_assemble_Q<<<(NBATCH * MDIM * MDIM + 255) / 256, 256, 0, stream>>>(corr, alpha_d, reg, Q);
  k_pm<<<NBATCH * C_IN * 25, 256, 0, stream>>>(out, y, d, alpha_d, reg, Pm);
  k_cholesky_solve<<<NBATCH, 1024, 0, stream>>>(Q, Pm, out);
}